// Oracle_NN_69707319214090
// MI455X (gfx1250) — compile-verified
//
#include <hip/hip_runtime.h>
#include <math.h>

// CDNA5 (gfx1250) Chamfer-NN retrieval.
// d[m][n] = |x_m|^2 + |y_n|^2 - 2 x_m.y_n computed as ONE
// V_WMMA_F32_16X16X4_F32 per 16x16 tile:
//   A row (16x4)  = (x0, x1, x2, |x|^2)      -> lanes lo: K0,K1 ; hi: K2,K3
//   B col (4x16)  = (-2y0, -2y1, -2y2, 1)    -> lanes lo: K0,K1 ; hi: K2,K3
//   C  (16x16)    = |y_n|^2  (column-only -> per-lane register splat)
// Inner loop: WMMA + exactly one v_min3 consumption per result element
// (row tree + column tree each take 2 new values per op). Reductions are
// emitted via inline asm to bypass LLVM's minnum canonicalization, which
// otherwise inserts one v_max_num_f32 x,x,x per WMMA output element.
//
// Assumed 32-bit WMMA operand layouts (ISA 7.12.2):
//   A 16x4:  VGPR0 = K0 (lanes 0-15) | K2 (lanes 16-31); VGPR1 = K1 | K3
//   B 4x16:  mirrored; C/D: VGPR v, lane l -> (M = v + 8*(l>=16), N = l%16)

typedef __attribute__((ext_vector_type(2))) float v2f;
typedef __attribute__((ext_vector_type(8))) float v8f;

#define CH_EPS 1e-12f
#define NPTS   1024
#define NTRAIN 128
#define NTEST  8

// IEEE minimumNumber/maximumNumber ops without compiler-inserted input
// canonicalization (inputs are finite products of finite data).
__device__ __forceinline__ float fmin3_raw(float a, float b, float c) {
    float d;
    asm("v_min3_num_f32 %0, %1, %2, %3" : "=v"(d) : "v"(a), "v"(b), "v"(c));
    return d;
}
__device__ __forceinline__ float fmin2_raw(float a, float b) {
    float d;
    asm("v_min_num_f32 %0, %1, %2" : "=v"(d) : "v"(a), "v"(b));
    return d;
}
__device__ __forceinline__ float clamp0_raw(float a) {
    float d;  // max(0, a): inline constant must be src0 in VOP2
    asm("v_max_num_f32 %0, 0, %1" : "=v"(d) : "v"(a));
    return d;
}

__global__ __launch_bounds__(256) void chamfer_pair_kernel(
    const float* __restrict__ train, const float* __restrict__ test,
    float* __restrict__ losses)
{
    __shared__ float4 Xs[NPTS];      // test:  (x0, x1, x2, |x|^2)
    __shared__ float4 Ys[NPTS];      // train: (-2y0, -2y1, -2y2, |y|^2)
    __shared__ int    rowminI[NPTS]; // float bits, values clamped >= 0
    __shared__ int    colminI[NPTS];
    __shared__ float  part1[8], part2[8];

    const int tid = threadIdx.x;
    const int b = blockIdx.x >> 7;   // test cloud index (0..7)
    const int t = blockIdx.x & 127;  // train cloud index (0..127)
    const float* __restrict__ X = test  + b * (NPTS * 3);
    const float* __restrict__ Y = train + t * (NPTS * 3);

    for (int p = tid; p < NPTS; p += 256) {
        float x0 = X[3*p+0], x1 = X[3*p+1], x2 = X[3*p+2];
        Xs[p] = make_float4(x0, x1, x2, x0*x0 + x1*x1 + x2*x2);
        float y0 = Y[3*p+0], y1 = Y[3*p+1], y2 = Y[3*p+2];
        Ys[p] = make_float4(-2.0f*y0, -2.0f*y1, -2.0f*y2, y0*y0 + y1*y1 + y2*y2);
        rowminI[p] = 0x7F800000;  // +inf
        colminI[p] = 0x7F800000;
    }
    __syncthreads();

    const int  wave = tid >> 5;
    const int  lane = tid & 31;
    const bool hi   = lane >= 16;
    const int  ln   = lane & 15;

    // Resident A tiles for this wave's 8 tile-rows.
    v2f a[8];
    #pragma unroll
    for (int rr = 0; rr < 8; ++rr) {
        float4 xm = Xs[(wave * 8 + rr) * 16 + ln];
        a[rr].x = hi ? xm.z : xm.x;   // K0 | K2
        a[rr].y = hi ? xm.w : xm.y;   // K1 | K3 (= |x|^2)
    }

    // Row-min accumulators (elementwise over column tiles; cross-lane at end).
    float racc[8][8];
    #pragma unroll
    for (int rr = 0; rr < 8; ++rr)
        #pragma unroll
        for (int v = 0; v < 8; ++v) racc[rr][v] = INFINITY;

    // Column tiles processed in pairs; start staggered per wave to spread
    // the colmin atomics across addresses.
    for (int jj = 0; jj < 32; ++jj) {
        const int j0 = ((jj * 2) + wave * 8) & 63;
        const int n0 = j0 * 16 + ln;
        const int n1 = n0 + 16;
        float4 y0 = Ys[n0];
        float4 y1 = Ys[n1];
        v2f b0, b1;
        b0.x = hi ? y0.z : y0.x;   b0.y = hi ? 1.0f : y0.y;
        b1.x = hi ? y1.z : y1.x;   b1.y = hi ? 1.0f : y1.y;
        v8f cs0, cs1;              // C[m][n] = |y_n|^2 -> lane splat
        #pragma unroll
        for (int v = 0; v < 8; ++v) { cs0[v] = y0.w; cs1[v] = y1.w; }

        float c0[8], c1[8];        // per-pair column accumulators
        #pragma unroll
        for (int v = 0; v < 8; ++v) { c0[v] = INFINITY; c1[v] = INFINITY; }

        #pragma unroll
        for (int rp = 0; rp < 4; ++rp) {
            v8f d00 = __builtin_amdgcn_wmma_f32_16x16x4_f32(
                false, a[2*rp+0], false, b0, (short)0, cs0, false, false);
            v8f d01 = __builtin_amdgcn_wmma_f32_16x16x4_f32(
                false, a[2*rp+0], false, b1, (short)0, cs1, false, false);
            v8f d10 = __builtin_amdgcn_wmma_f32_16x16x4_f32(
                false, a[2*rp+1], false, b0, (short)0, cs0, false, false);
            v8f d11 = __builtin_amdgcn_wmma_f32_16x16x4_f32(
                false, a[2*rp+1], false, b1, (short)0, cs1, false, false);
            #pragma unroll
            for (int v = 0; v < 8; ++v) {
                // row accumulation: merge the two column tiles
                racc[2*rp+0][v] = fmin3_raw(racc[2*rp+0][v], d00[v], d01[v]);
                racc[2*rp+1][v] = fmin3_raw(racc[2*rp+1][v], d10[v], d11[v]);
                // column accumulation: merge the two tile-rows
                c0[v] = fmin3_raw(c0[v], d00[v], d10[v]);
                c1[v] = fmin3_raw(c1[v], d01[v], d11[v]);
            }
        }
        float m0 = fmin2_raw(fmin2_raw(fmin3_raw(c0[0], c0[1], c0[2]),
                                       fmin3_raw(c0[3], c0[4], c0[5])),
                             fmin2_raw(c0[6], c0[7]));
        float m1 = fmin2_raw(fmin2_raw(fmin3_raw(c1[0], c1[1], c1[2]),
                                       fmin3_raw(c1[3], c1[4], c1[5])),
                             fmin2_raw(c1[6], c1[7]));
        // lanes l and l+16 hit the same address: ds_min_i32 merges the halves
        atomicMin(&colminI[n0], __float_as_int(clamp0_raw(m0)));
        atomicMin(&colminI[n1], __float_as_int(clamp0_raw(m1)));
    }

    // Row mins: cross-lane merge via LDS atomic (rows owned by this wave,
    // 16 lanes per half fold into one address each).
    #pragma unroll
    for (int rr = 0; rr < 8; ++rr) {
        const int rowbase = (wave * 8 + rr) * 16;
        #pragma unroll
        for (int v = 0; v < 8; ++v)
            atomicMin(&rowminI[rowbase + v + (hi ? 8 : 0)],
                      __float_as_int(clamp0_raw(racc[rr][v])));
    }
    __syncthreads();

    // loss = 0.5 * (mean_i sqrt(rowmin_i+eps) + mean_j sqrt(colmin_j+eps))
    float s1 = 0.f, s2 = 0.f;
    for (int p = tid; p < NPTS; p += 256) {
        s1 += sqrtf(__int_as_float(rowminI[p]) + CH_EPS);
        s2 += sqrtf(__int_as_float(colminI[p]) + CH_EPS);
    }
    #pragma unroll
    for (int off = 16; off >= 1; off >>= 1) {
        s1 += __shfl_xor(s1, off, 32);
        s2 += __shfl_xor(s2, off, 32);
    }
    if (lane == 0) { part1[wave] = s1; part2[wave] = s2; }
    __syncthreads();
    if (tid == 0) {
        float t1 = 0.f, t2 = 0.f;
        for (int w = 0; w < 8; ++w) { t1 += part1[w]; t2 += part2[w]; }
        losses[b * NTRAIN + t] = 0.5f * (t1 + t2) * (1.0f / NPTS);
    }
}

// Per test cloud: first-argmin over 128 train losses, gather winning cloud.
__global__ __launch_bounds__(128) void argmin_copy_kernel(
    const float* __restrict__ losses, const float* __restrict__ train,
    float* __restrict__ out, float* __restrict__ minloss)
{
    __shared__ float vv[128];
    __shared__ int   ix[128];
    const int b = blockIdx.x;
    const int t = threadIdx.x;
    vv[t] = losses[b * NTRAIN + t];
    ix[t] = t;
    __syncthreads();
    for (int s = 64; s > 0; s >>= 1) {
        if (t < s) {
            float vo = vv[t + s]; int io = ix[t + s];
            // strict '<' with index tie-break == JAX first-argmin
            if (vo < vv[t] || (vo == vv[t] && io < ix[t])) { vv[t] = vo; ix[t] = io; }
        }
        __syncthreads();
    }
    if (t == 0) minloss[b] = vv[0];
    const int idx = ix[0];
    const float* __restrict__ src = train + idx * (NPTS * 3);
    float* __restrict__ dst = out + 1 + b * (NPTS * 3);
    for (int k = t; k < NPTS * 3; k += 128) dst[k] = src[k];
}

__global__ void mean_kernel(const float* __restrict__ minloss,
                            float* __restrict__ out)
{
    if (threadIdx.x == 0) {
        float s = 0.f;
        for (int i = 0; i < NTEST; ++i) s += minloss[i];
        out[0] = s * (1.0f / NTEST);
    }
}

extern "C" void kernel_launch(void* const* d_in, const int* in_sizes, int n_in,
                              void* d_out, int out_size, void* d_ws, size_t ws_size,
                              hipStream_t stream)
{
    const float* train = (const float*)d_in[0];  // [128,1024,3]
    const float* test  = (const float*)d_in[1];  // [8,1024,3]
    float* out    = (float*)d_out;               // [1 + 8*1024*3]
    float* losses = (float*)d_ws;                // [8*128]
    float* minls  = losses + NTEST * NTRAIN;     // [8]

    chamfer_pair_kernel<<<dim3(NTEST * NTRAIN), dim3(256), 0, stream>>>(train, test, losses);
    argmin_copy_kernel<<<dim3(NTEST), dim3(128), 0, stream>>>(losses, train, out, minls);
    mean_kernel<<<dim3(1), dim3(32), 0, stream>>>(minls, out);
}